// EVGQueryAndGroup_48215302865116
// MI455X (gfx1250) — compile-verified
//
#include <hip/hip_runtime.h>

// ---------------------------------------------------------------------------
// EVG query-and-group for MI455X (gfx1250, wave32).
//   xyz:      (2, 16384, 3) f32
//   new_xyz:  (2,  2048, 3) f32
//   features: (2,    64, 16384) f32
//   out:      (2, 67, 2048, 32) f32
// Kernel 1: TDM-streamed point chunks -> WMMA distance tiles -> top-32 ->
//           covariance -> power iteration -> capsule endpoints (in d_ws).
// Kernel 2: capsule query (first-32 in index order) -> grouped gather.
// ---------------------------------------------------------------------------

typedef __attribute__((ext_vector_type(2))) float v2f;
typedef __attribute__((ext_vector_type(8))) float v8f;
typedef __attribute__((ext_vector_type(4))) unsigned int u32x4;
typedef __attribute__((ext_vector_type(4))) int i32x4;
typedef __attribute__((ext_vector_type(8))) int i32x8;

constexpr int NB    = 2;      // batches
constexpr int NPTS  = 16384;  // cloud points
constexpr int NQ    = 2048;   // query points
constexpr int NCH   = 64;     // feature channels
constexpr int KNN   = 32;     // knn nsample
constexpr int NS    = 32;     // vector-query nsample
constexpr int QT    = 16;     // queries per workgroup (kernel 1)
constexpr int CHUNK = 256;    // points per streamed chunk
constexpr int NCHK  = NPTS / CHUNK;
constexpr float RADIUS = 0.1f;
constexpr float VLEN   = 0.2f;
constexpr int PITERS   = 20;

// ---------------------------------------------------------------------------
// TDM: DMA one contiguous chunk of CHUNK*3 f32 elements from global to LDS.
// D# per CDNA5 ISA:
//   group0: count=1 | lds_addr | global_addr[56:0] | type=2
//   group1: data_size=4B, tensor = 768x1, tile = 768x1, strides = 768
// Issued by one wave; tracked with TENSORcnt.
// ---------------------------------------------------------------------------
__device__ __forceinline__ void tdm_load_chunk(const float* gsrc, float* lds_dst)
{
    const unsigned long long ga = (unsigned long long)gsrc;
    const unsigned lds_off = (unsigned)(unsigned long long)lds_dst; // aperture low 32 = LDS offset
    constexpr unsigned ELEMS = (unsigned)(CHUNK * 3);

    u32x4 g0;
    g0.x = 1u;                                              // count=1 (valid user D#)
    g0.y = lds_off;                                         // lds_addr (bytes)
    g0.z = (unsigned)(ga & 0xFFFFFFFFull);                  // global_addr[31:0]
    g0.w = (unsigned)((ga >> 32) & 0x01FFFFFFull) | (2u << 30); // addr[56:32] | type=2

    i32x8 g1;
    g1[0] = (int)(2u << 16);                 // data_size = 2 (4 bytes)
    g1[1] = (int)((ELEMS & 0xFFFFu) << 16);  // tensor_dim0[15:0]
    g1[2] = (int)((ELEMS >> 16) | (1u << 16)); // tensor_dim0[31:16] | tensor_dim1[15:0]=1
    g1[3] = (int)(ELEMS << 16);              // tensor_dim1[31:16]=0 | tile_dim0
    g1[4] = 1;                               // tile_dim1=1, tile_dim2=0
    g1[5] = (int)ELEMS;                      // tensor_dim0_stride[31:0]
    g1[6] = (int)((ELEMS & 0xFFFFu) << 16);  // stride0[47:32]=0 | stride1[15:0]
    g1[7] = 0;                               // stride1[47:16]=0

    i32x4 gz = {0, 0, 0, 0};
#if __clang_major__ >= 23
    i32x8 gz8 = {0, 0, 0, 0, 0, 0, 0, 0};
    __builtin_amdgcn_tensor_load_to_lds(g0, g1, gz, gz, gz8, 0);
#else
    __builtin_amdgcn_tensor_load_to_lds(g0, g1, gz, gz, 0);
#endif
}

// ===========================================================================
// Kernel 1: per tile of 16 queries, stream all points through LDS via TDM
// (double-buffered), build 16x256 distance tiles with V_WMMA_F32_16X16X4_F32:
//   row_m = (-2qx, -2qy, -2qz, 1),  col_n = (x, y, z, |x|^2)
//   => D[m][n] = |x_n|^2 - 2 q_m . x_n   (= d^2 up to per-row const |q|^2)
// then maintain a 32-entry replace-max heap per query in LDS.
// ===========================================================================
__global__ __launch_bounds__(256)
void evg_knn_dirs_kernel(const float* __restrict__ xyz,
                         const float* __restrict__ new_xyz,
                         float* __restrict__ p1p2)
{
    __shared__ float sRaw[2][CHUNK * 3];   // TDM double buffer
    __shared__ float sQ[QT][3];
    __shared__ float sDist[QT][CHUNK];
    __shared__ float sHD[QT][KNN];
    __shared__ int   sHI[QT][KNN];
    __shared__ float sHMax[QT];
    __shared__ int   sHArg[QT];

    const int tid  = threadIdx.x;
    const int lane = tid & 31;
    const int wid  = tid >> 5;
    const int bq   = blockIdx.x;               // 0 .. NB*(NQ/QT)-1
    const int b    = bq / (NQ / QT);
    const int q0   = (bq % (NQ / QT)) * QT;
    const float* gxyz = xyz + (size_t)b * NPTS * 3;

    if (tid < QT * 3) {
        int m = tid / 3, c = tid % 3;
        sQ[m][c] = new_xyz[(b * NQ + q0 + m) * 3 + c];
    }
    if (tid < QT) { sHMax[tid] = 3.4e38f; sHArg[tid] = 0; }
    for (int e = tid; e < QT * KNN; e += 256) {
        (&sHD[0][0])[e] = 3.4e38f;
        (&sHI[0][0])[e] = 0;
    }

    // Kick off the first chunk DMA (wave 0 only; TDM ignores EXEC).
    if (wid == 0) tdm_load_chunk(gxyz, &sRaw[0][0]);
    __syncthreads();

    // A operand (16x4 f32 ISA layout: lanes 0-15 hold K=0,1; lanes 16-31 K=2,3)
    v2f A;
    {
        const int m = lane & 15;
        if (lane < 16) { A.x = -2.0f * sQ[m][0]; A.y = -2.0f * sQ[m][1]; }
        else           { A.x = -2.0f * sQ[m][2]; A.y = 1.0f; }
    }

    for (int ch = 0; ch < NCHK; ++ch) {
        const int base = ch * CHUNK;
        // Issue next chunk, then wait until only it is outstanding -> chunk ch done.
        if (wid == 0) {
            if (ch + 1 < NCHK) {
                tdm_load_chunk(gxyz + (size_t)(base + CHUNK) * 3,
                               &sRaw[(ch + 1) & 1][0]);
                __builtin_amdgcn_s_wait_tensorcnt(1);
            } else {
                __builtin_amdgcn_s_wait_tensorcnt(0);
            }
        }
        __syncthreads();
        const float* raw = &sRaw[ch & 1][0];

        // Each wave computes two 16(query) x 16(point) WMMA sub-tiles.
        for (int sub = 0; sub < 2; ++sub) {
            const int nb = (wid * 2 + sub) * 16;
            const int n  = lane & 15;
            const int pi = (nb + n) * 3;
            const float px = raw[pi + 0];
            const float py = raw[pi + 1];
            const float pz = raw[pi + 2];
            v2f Bv;
            if (lane < 16) { Bv.x = px; Bv.y = py; }
            else           { Bv.x = pz; Bv.y = px * px + py * py + pz * pz; }
            v8f Cv = {};
            v8f D = __builtin_amdgcn_wmma_f32_16x16x4_f32(
                false, A, false, Bv, (short)0, Cv, false, false);
            // C/D layout: VGPR r -> (M=r, lanes 0-15) / (M=r+8, lanes 16-31)
            const int row0 = (lane < 16) ? 0 : 8;
            const int col  = nb + (lane & 15);
#pragma unroll
            for (int r = 0; r < 8; ++r)
                sDist[row0 + r][col] = D[r];
        }
        __syncthreads();

        // Top-32 maintenance: wave w serves queries w and w+8.
        for (int pass = 0; pass < 2; ++pass) {
            const int q = wid + pass * 8;
#pragma unroll
            for (int j = 0; j < CHUNK / 32; ++j) {
                const int   col = j * 32 + lane;
                const float dv  = sDist[q][col];
                const int   gi  = base + col;
                bool want = dv < sHMax[q];
                while (__any(want)) {
                    const unsigned mb = (unsigned)__ballot(want);
                    const int leader  = __ffs(mb) - 1;
                    if (lane == leader) {
                        const int a = sHArg[q];
                        sHD[q][a] = dv; sHI[q][a] = gi;
                        float mx = -3.4e38f; int am = 0;
                        for (int e = 0; e < KNN; ++e)
                            if (sHD[q][e] > mx) { mx = sHD[q][e]; am = e; }
                        sHMax[q] = mx; sHArg[q] = am;
                        want = false;
                    }
                    want = want && (dv < sHMax[q]);
                }
            }
        }
        __syncthreads();
    }

    // Covariance over the 32 NN (one neighbor per lane) + power iteration.
    for (int pass = 0; pass < 2; ++pass) {
        const int q   = wid + pass * 8;
        const int idx = sHI[q][lane];
        const float x = gxyz[idx * 3 + 0];
        const float y = gxyz[idx * 3 + 1];
        const float z = gxyz[idx * 3 + 2];
        float mx = x, my = y, mz = z;
        for (int o = 16; o > 0; o >>= 1) {
            mx += __shfl_xor(mx, o, 32);
            my += __shfl_xor(my, o, 32);
            mz += __shfl_xor(mz, o, 32);
        }
        mx *= (1.0f / 32.0f); my *= (1.0f / 32.0f); mz *= (1.0f / 32.0f);
        const float cx = x - mx, cy = y - my, cz = z - mz;
        float xx = cx * cx, xy = cx * cy, xz = cx * cz;
        float yy = cy * cy, yz = cy * cz, zz = cz * cz;
        for (int o = 16; o > 0; o >>= 1) {
            xx += __shfl_xor(xx, o, 32); xy += __shfl_xor(xy, o, 32);
            xz += __shfl_xor(xz, o, 32); yy += __shfl_xor(yy, o, 32);
            yz += __shfl_xor(yz, o, 32); zz += __shfl_xor(zz, o, 32);
        }
        const float s = 1.0f / 31.0f;
        xx *= s; xy *= s; xz *= s; yy *= s; yz *= s; zz *= s;
        float vx = 1.0f, vy = 1.0f, vz = 1.0f;
        for (int it = 0; it < PITERS; ++it) {
            const float tx = xx * vx + xy * vy + xz * vz;
            const float ty = xy * vx + yy * vy + yz * vz;
            const float tz = xz * vx + yz * vy + zz * vz;
            const float inv = 1.0f / sqrtf(tx * tx + ty * ty + tz * tz);
            vx = tx * inv; vy = ty * inv; vz = tz * inv;
        }
        if (lane == 0) {
            const int gp = b * NQ + q0 + q;
            float* o = p1p2 + (size_t)gp * 6;
            o[0] = sQ[q][0] - vx * VLEN; o[1] = sQ[q][1] - vy * VLEN;
            o[2] = sQ[q][2] - vz * VLEN;
            o[3] = sQ[q][0] + vx * VLEN; o[4] = sQ[q][1] + vy * VLEN;
            o[5] = sQ[q][2] + vz * VLEN;
        }
    }
}

// ===========================================================================
// Kernel 2: one workgroup per query point. In-order chunked scan for the
// first 32 points within RADIUS of segment p1->p2 (ballot + popc prefix
// compaction keeps index order), then gather 3 coord + 64 feature channels.
// ===========================================================================
__global__ __launch_bounds__(256)
void evg_capsule_group_kernel(const float* __restrict__ xyz,
                              const float* __restrict__ new_xyz,
                              const float* __restrict__ feat,
                              const float* __restrict__ p1p2,
                              float* __restrict__ out)
{
    __shared__ int sIdx[NS];
    __shared__ int sWCnt[8];
    __shared__ int sCnt;

    const int tid  = threadIdx.x;
    const int lane = tid & 31;
    const int wid  = tid >> 5;
    const int gp   = blockIdx.x;       // 0 .. NB*NQ-1
    const int b    = gp / NQ;
    const int p    = gp % NQ;

    const float p1x = p1p2[gp * 6 + 0];
    const float p1y = p1p2[gp * 6 + 1];
    const float p1z = p1p2[gp * 6 + 2];
    const float dx  = p1p2[gp * 6 + 3] - p1x;
    const float dy  = p1p2[gp * 6 + 4] - p1y;
    const float dz  = p1p2[gp * 6 + 5] - p1z;
    const float inv_dd = 1.0f / fmaxf(dx * dx + dy * dy + dz * dz, 1e-12f);
    const float r2 = RADIUS * RADIUS;

    if (tid == 0) sCnt = 0;
    __syncthreads();

    for (int ch = 0; ch < NPTS / 256; ++ch) {
        const int   i = ch * 256 + tid;
        if (ch + 1 < NPTS / 256)     // global_prefetch_b8 for the next chunk
            __builtin_prefetch(&xyz[(b * NPTS + i + 256) * 3], 0, 1);
        const float x = xyz[(b * NPTS + i) * 3 + 0];
        const float y = xyz[(b * NPTS + i) * 3 + 1];
        const float z = xyz[(b * NPTS + i) * 3 + 2];
        const float rx = x - p1x, ry = y - p1y, rz = z - p1z;
        float t = (rx * dx + ry * dy + rz * dz) * inv_dd;
        t = fminf(fmaxf(t, 0.0f), 1.0f);
        const float ex = rx - t * dx, ey = ry - t * dy, ez = rz - t * dz;
        const bool hit = (ex * ex + ey * ey + ez * ez) <= r2;

        const unsigned mb = (unsigned)__ballot(hit);
        if (lane == 0) sWCnt[wid] = __popc(mb);
        __syncthreads();
        int off = sCnt;
        for (int w = 0; w < wid; ++w) off += sWCnt[w];
        int tot = 0;
        for (int w = 0; w < 8; ++w) tot += sWCnt[w];
        const int pos = off + __popc(mb & ((1u << lane) - 1u));
        if (hit && pos < NS) sIdx[pos] = i;
        __syncthreads();
        if (tid == 0) sCnt += tot;
        __syncthreads();
        if (sCnt >= NS) break;      // uniform across the workgroup
    }

    // Padding: empty -> N-1 everywhere; else replicate first hit.
    const int cnt = (sCnt < NS) ? sCnt : NS;
    if (tid < NS) {
        if (cnt == 0)        sIdx[tid] = NPTS - 1;
        else if (tid >= cnt) sIdx[tid] = sIdx[0];
    }
    __syncthreads();

    const float cqx = new_xyz[gp * 3 + 0];
    const float cqy = new_xyz[gp * 3 + 1];
    const float cqz = new_xyz[gp * 3 + 2];

    // out: (NB, 3+NCH, NQ, NS)
    for (int e = tid; e < (NCH + 3) * NS; e += 256) {
        const int c = e >> 5;
        const int s = e & 31;
        const int idx = sIdx[s];
        float v;
        if (c < 3)
            v = xyz[(b * NPTS + idx) * 3 + c] - (c == 0 ? cqx : (c == 1 ? cqy : cqz));
        else
            v = feat[((size_t)b * NCH + (c - 3)) * NPTS + idx];
        out[(((size_t)b * (NCH + 3) + c) * NQ + p) * NS + s] = v;
    }
}

// ===========================================================================
extern "C" void kernel_launch(void* const* d_in, const int* in_sizes, int n_in,
                              void* d_out, int out_size, void* d_ws, size_t ws_size,
                              hipStream_t stream)
{
    const float* xyz     = (const float*)d_in[0];
    const float* new_xyz = (const float*)d_in[1];
    const float* feat    = (const float*)d_in[2];
    float* p1p2 = (float*)d_ws;                 // NB*NQ*6 floats = 96 KB scratch

    evg_knn_dirs_kernel<<<NB * (NQ / QT), 256, 0, stream>>>(xyz, new_xyz, p1p2);
    evg_capsule_group_kernel<<<NB * NQ, 256, 0, stream>>>(xyz, new_xyz, feat,
                                                          p1p2, (float*)d_out);
}